// LSTMModel_84172769067192
// MI455X (gfx1250) — compile-verified
//
#include <hip/hip_runtime.h>

typedef float v2f __attribute__((ext_vector_type(2)));
typedef float v8f __attribute__((ext_vector_type(8)));

#define Bsz 2048
#define Tsz 1024
#define Isz 8
#define Hsz 64
#define NKv 18      // K = 72 (= H 64 + I 8) in steps of 4
#define ASTR 74     // padded LDS row stride (even -> 8B aligned b64 loads, conflict-free)

__device__ __forceinline__ float sigmoid_f(float v) {
    return 1.0f / (1.0f + __expf(-v));
}
__device__ __forceinline__ float tanh_f(float v) {
    // NaN-safe fast tanh: exp(-2|v|) in [0,1]
    float e = __expf(-2.0f * fabsf(v));
    float r = (1.0f - e) / (1.0f + e);
    return copysignf(r, v);
}
__device__ __forceinline__ v8f splat8(float v) {
    v8f r;
#pragma unroll
    for (int i = 0; i < 8; ++i) r[i] = v;
    return r;
}

// One workgroup = 16 batch rows, 4 waves. Wave w owns hidden cols [16w,16w+16)
// across all 4 gates, so the c-state update is wave-local; h is exchanged via LDS.
__global__ __launch_bounds__(128, 1) void lstm_fused_kernel(
    const float* __restrict__ x, const float* __restrict__ W_ih,
    const float* __restrict__ W_hh, const float* __restrict__ b_ih,
    const float* __restrict__ b_hh, const float* __restrict__ fc_w,
    const float* __restrict__ fc_b, float* __restrict__ out)
{
    __shared__ float As[16 * ASTR];   // A tile: [h (64) | x_t (8)] per batch row, padded

    const int tid   = threadIdx.x;
    const int lane  = tid & 31;
    const int wave  = tid >> 5;       // 0..3 -> hidden slice base n0 = 16*wave
    const int n16   = lane & 15;
    const int kHalf = lane >> 4;      // 0/1 (WMMA f32 A/B lane split over K)
    const int bBase = blockIdx.x * 16;

    // ---- preload B operand (combined [W_hh^T ; W_ih^T], 72 x 256) into VGPRs ----
    // B 4x16 layout: lanes 0-15 hold rows K=0(v0),K=1(v1); lanes 16-31 rows K=2,K=3.
    v2f Bfrag[4][NKv];
#pragma unroll
    for (int q = 0; q < 4; ++q) {
        const int g = q * 64 + wave * 16 + n16;   // gate output index 0..255
#pragma unroll
        for (int kk = 0; kk < NKv; ++kk) {
            const int k0 = kk * 4 + 2 * kHalf;
            v2f f;
            if (kk < 16) {            // recurrent part: Wc[k][g] = W_hh[g][k]
                f.x = W_hh[g * Hsz + k0];
                f.y = W_hh[g * Hsz + k0 + 1];
            } else {                  // input part: Wc[64+i][g] = W_ih[g][i]
                f.x = W_ih[g * Isz + (k0 - 64)];
                f.y = W_ih[g * Isz + (k0 - 63)];
            }
            Bfrag[q][kk] = f;
        }
    }

    float bias[4];
#pragma unroll
    for (int q = 0; q < 4; ++q) {
        const int g = q * 64 + wave * 16 + n16;
        bias[q] = b_ih[g] + b_hh[g];
    }

    // h = 0, then stage x_0 into A-tile columns 64..71
    for (int idx = tid; idx < 16 * ASTR; idx += 128) As[idx] = 0.0f;
    __syncthreads();
    const int xm = tid >> 3;   // batch row 0..15
    const int xi = tid & 7;    // input feature 0..7
    As[xm * ASTR + 64 + xi] = x[((size_t)(bBase + xm) * Tsz + 0) * Isz + xi];
    __syncthreads();

    v8f c = splat8(0.0f);      // cell state, wave-local, WMMA D layout

    for (int t = 0; t < Tsz; ++t) {
        v8f acc0 = splat8(bias[0]);   // i
        v8f acc1 = splat8(bias[1]);   // f
        v8f acc2 = splat8(bias[2]);   // g
        v8f acc3 = splat8(bias[3]);   // o
#pragma unroll
        for (int kk = 0; kk < NKv; ++kk) {
            // A 16x4 layout: lanes 0-15: M=lane, v0=K0,v1=K1; lanes 16-31: M=lane-16, v0=K2,v1=K3
            const v2f a = *(const v2f*)&As[n16 * ASTR + kk * 4 + 2 * kHalf];
            acc0 = __builtin_amdgcn_wmma_f32_16x16x4_f32(false, a, false, Bfrag[0][kk],
                                                         (short)0, acc0, false, false);
            acc1 = __builtin_amdgcn_wmma_f32_16x16x4_f32(false, a, false, Bfrag[1][kk],
                                                         (short)0, acc1, false, false);
            acc2 = __builtin_amdgcn_wmma_f32_16x16x4_f32(false, a, false, Bfrag[2][kk],
                                                         (short)0, acc2, false, false);
            acc3 = __builtin_amdgcn_wmma_f32_16x16x4_f32(false, a, false, Bfrag[3][kk],
                                                         (short)0, acc3, false, false);
        }

        // prefetch next timestep's x element (uniform condition, no divergence)
        float xn = 0.0f;
        if (t + 1 < Tsz)
            xn = x[((size_t)(bBase + xm) * Tsz + (t + 1)) * Isz + xi];

        // wave-local LSTM cell update in D layout
        v8f hnew;
#pragma unroll
        for (int r = 0; r < 8; ++r) {
            float ig = sigmoid_f(acc0[r]);
            float fg = sigmoid_f(acc1[r]);
            float gg = tanh_f(acc2[r]);
            float og = sigmoid_f(acc3[r]);
            float cn = fg * c[r] + ig * gg;
            c[r]     = cn;
            hnew[r]  = og * tanh_f(cn);
        }

        __syncthreads();   // all waves finished reading As for step t
        // D layout: VGPR r -> row r (lanes 0-15) / row r+8 (lanes 16-31), col = n0 + n16
#pragma unroll
        for (int r = 0; r < 8; ++r) {
            const int mrow = r + 8 * kHalf;
            As[mrow * ASTR + wave * 16 + n16] = hnew[r];
        }
        As[xm * ASTR + 64 + xi] = xn;
        __syncthreads();   // h_{t+1}, x_{t+1} visible
    }

    // out = sigmoid(hT @ fc_w^T + fc_b), one lane per batch row
    if (tid < 16) {
        float s = fc_b[0];
#pragma unroll 8
        for (int k = 0; k < Hsz; ++k) s += As[tid * ASTR + k] * fc_w[k];
        out[bBase + tid] = sigmoid_f(s);
    }
}

extern "C" void kernel_launch(void* const* d_in, const int* in_sizes, int n_in,
                              void* d_out, int out_size, void* d_ws, size_t ws_size,
                              hipStream_t stream) {
    (void)in_sizes; (void)n_in; (void)d_ws; (void)ws_size; (void)out_size;
    const float* x    = (const float*)d_in[0];
    const float* W_ih = (const float*)d_in[1];
    const float* W_hh = (const float*)d_in[2];
    const float* b_ih = (const float*)d_in[3];
    const float* b_hh = (const float*)d_in[4];
    const float* fc_w = (const float*)d_in[5];
    const float* fc_b = (const float*)d_in[6];
    float* out = (float*)d_out;

    dim3 grid(Bsz / 16);   // 128 workgroups
    dim3 block(128);       // 4 waves of 32
    lstm_fused_kernel<<<grid, block, 0, stream>>>(x, W_ih, W_hh, b_ih, b_hh,
                                                  fc_w, fc_b, out);
}